// QuantumEntanglement_45183055954392
// MI455X (gfx1250) — compile-verified
//
#include <hip/hip_runtime.h>
#include <hip/hip_bf16.h>
#include <math.h>

typedef __attribute__((ext_vector_type(2))) float v2f;
typedef __attribute__((ext_vector_type(8))) float v8f;

#define BATCH   16384
#define NE      8
#define DM      512
#define NROWS   (BATCH * NE)     // 131072 flattened expert rows
#define NGROUPS (NROWS / 16)     // 8192 groups of 16 rows (= 2 batches each)
#define NCTILES (DM / 16)        // 32 column tiles of 16

// ---------------------------------------------------------------------------
// Kernel 1: A[i][j] = (i==j) ? 1 : sigmoid(corr[i][j])   (the I + W matrix)
// ---------------------------------------------------------------------------
__global__ void qe_prep_A(const float* __restrict__ corr, float* __restrict__ A8) {
    int t = threadIdx.x;
    if (t < NE * NE) {
        int i = t >> 3, j = t & 7;
        float c = corr[t];
        float s = 1.0f / (1.0f + __expf(-c));
        A8[t] = (i == j) ? 1.0f : s;
    }
}

// ---------------------------------------------------------------------------
// Kernel 2: streaming entangle via v_wmma_f32_16x16x4_f32.
// One wave computes one 16x16 output tile:
//   rows  = 16 consecutive flattened rows (2 batches x 8 experts)
//   cols  = 16 consecutive d_model columns
//   D     = sum_{kc=0..3} A16[:, 4kc:4kc+4] x X[16g + 4kc : ..., n0:n0+16]
// where A16 = blockdiag(A8, A8).
// ---------------------------------------------------------------------------
__global__ __launch_bounds__(256) void qe_entangle_wmma(const float* __restrict__ x,
                                                        const float* __restrict__ A8,
                                                        float* __restrict__ out) {
    const int lane = threadIdx.x & 31;
    const int half = lane >> 4;      // 0: lanes 0-15, 1: lanes 16-31
    const int l    = lane & 15;
    const int wave = (int)((blockIdx.x * blockDim.x + threadIdx.x) >> 5);
    const int g    = wave >> 5;          // row-group index (wave-uniform)
    const int n0   = (wave & 31) << 4;   // column-tile start
    if (g >= NGROUPS) return;            // never taken with exact grid (EXEC stays all-1s)

    // Per-lane A-matrix fragments for the 4 K-chunks.
    // A layout (16x4 f32): lane l (low half) holds A16[l][4kc+0..1] in v0..v1,
    // lanes 16-31 hold A16[l][4kc+2..3]. Block-diagonal: zero unless (row&8)==(col&8).
    float a[4][2];
#pragma unroll
    for (int kc = 0; kc < 4; ++kc) {
#pragma unroll
        for (int u = 0; u < 2; ++u) {
            int c = kc * 4 + 2 * half + u;
            a[kc][u] = ((l ^ c) & 8) ? 0.0f : A8[(l & 7) * 8 + (c & 7)];
        }
    }

    const float* xg = x + (size_t)(16 * g) * DM + n0 + l;
    v8f acc = {};
#pragma unroll
    for (int kc = 0; kc < 4; ++kc) {
        const int k0 = kc * 4 + 2 * half;
        v2f av; av.x = a[kc][0];               av.y = a[kc][1];
        v2f bv; bv.x = xg[(size_t)k0 * DM];    bv.y = xg[(size_t)(k0 + 1) * DM];
        // (neg_a, A, neg_b, B, c_mod, C, reuse_a, reuse_b)
        acc = __builtin_amdgcn_wmma_f32_16x16x4_f32(false, av, false, bv,
                                                    (short)0, acc, false, false);
    }

    // C/D layout: VGPR r -> row r (lanes 0-15) / row r+8 (lanes 16-31), col = lane%16
    float* og = out + (size_t)(16 * g) * DM + n0 + l;
#pragma unroll
    for (int r = 0; r < 8; ++r) {
        og[(size_t)(r + 8 * half) * DM] = acc[r];
    }
}

// ---------------------------------------------------------------------------
// Kernel 3: measure = |sum(entangled[0]^2)| / (E*D + 1e-8).
// entangled[0] is only 8x512 -> recompute from inputs, single workgroup.
// ---------------------------------------------------------------------------
__global__ __launch_bounds__(512) void qe_measure(const float* __restrict__ x,
                                                  const float* __restrict__ A8,
                                                  float* __restrict__ measure_out) {
    __shared__ float red[512];
    const int d = threadIdx.x;   // d_model column, batch 0
    float xv[NE];
#pragma unroll
    for (int j = 0; j < NE; ++j) xv[j] = x[(size_t)j * DM + d];
    float ss = 0.0f;
#pragma unroll
    for (int i = 0; i < NE; ++i) {
        float e = 0.0f;
#pragma unroll
        for (int j = 0; j < NE; ++j) e = fmaf(A8[i * 8 + j], xv[j], e);
        ss = fmaf(e, e, ss);
    }
    red[d] = ss;
    __syncthreads();
    for (int s = 256; s > 0; s >>= 1) {
        if (d < s) red[d] += red[d + s];
        __syncthreads();
    }
    if (d == 0) {
        measure_out[0] = fabsf(red[0]) / ((float)(NE * DM) + 1e-8f);
    }
}

// ---------------------------------------------------------------------------
extern "C" void kernel_launch(void* const* d_in, const int* in_sizes, int n_in,
                              void* d_out, int out_size, void* d_ws, size_t ws_size,
                              hipStream_t stream) {
    const float* x    = (const float*)d_in[0];   // (16384, 8, 512) f32
    const float* corr = (const float*)d_in[1];   // (8, 8) f32
    float* out        = (float*)d_out;           // entangled flat + 1 measure
    float* A8         = (float*)d_ws;            // 64 floats scratch

    qe_prep_A<<<1, 64, 0, stream>>>(corr, A8);

    // Exact grid: NGROUPS*NCTILES waves, 8 waves (256 threads) per block.
    const int total_waves = NGROUPS * NCTILES;           // 262144
    const int blocks      = total_waves / 8;             // 32768
    qe_entangle_wmma<<<blocks, 256, 0, stream>>>(x, A8, out);

    qe_measure<<<1, 512, 0, stream>>>(x, A8, out + (size_t)BATCH * NE * DM);
}